// OrthonormalTransform_10204842295824
// MI455X (gfx1250) — compile-verified
//
#include <hip/hip_runtime.h>

#define NPTS 64
#define NANG 2016
#define INV_2PI 0.15915494309189535f
#define LDS_STRIDE 66   // 64 + 2 pad: ds_load_b64 A-fragment reads hit distinct banks

typedef float v2f __attribute__((ext_vector_type(2)));
typedef float v8f __attribute__((ext_vector_type(8)));

// ---------------------------------------------------------------------------
// Kernel 1: fold the 2016 sequential Givens rotations (+ row scaling by mus)
// into a single 64x64 matrix  M = diag(mus) * R_2016 * ... * R_1.
//
// Phase A (parallel): 256 threads precompute all 2016 (cos,sin) pairs with
// hardware v_cos/v_sin (input pre-scaled by 1/2pi) into LDS.
// Phase B (serial chain): rotations only mix rows, so each column of M
// evolves independently -> threads 0..63 each own one column entirely in
// registers; the fully-unrolled loop is now only 2 LDS broadcast reads +
// 4 FMAs per rotation (trig is off the critical path).
// ---------------------------------------------------------------------------
__global__ __launch_bounds__(256) void fold_rotations_kernel(
    const float* __restrict__ angles,
    const float* __restrict__ mus,
    float* __restrict__ M)
{
  __shared__ float csh[NANG];
  __shared__ float ssh[NANG];

  const int tid = threadIdx.x;
  for (int k = tid; k < NANG; k += 256) {
    const float a = angles[k] * INV_2PI;
    csh[k] = __builtin_amdgcn_cosf(a);
    ssh[k] = __builtin_amdgcn_sinf(a);
  }
  __syncthreads();

  if (tid >= NPTS) return;
  const int j = tid;  // column of M owned by this thread

  float m[NPTS];
#pragma unroll
  for (int i = 0; i < NPTS; ++i) m[i] = (i == j) ? 1.0f : 0.0f;

  int k = 0;
#pragma unroll
  for (int it = 0; it < NPTS - 1; ++it) {
#pragma unroll
    for (int ib = it + 1; ib < NPTS; ++ib, ++k) {
      const float c = csh[k];           // LDS broadcast (uniform address)
      const float s = ssh[k];
      const float vt = m[it];
      const float vb = m[ib];
      m[it] = c * vt - s * vb;
      m[ib] = s * vt + c * vb;
    }
  }
#pragma unroll
  for (int i = 0; i < NPTS; ++i) M[i * NPTS + j] = m[i] * mus[i];
}

// ---------------------------------------------------------------------------
// Kernel 2: Y = M * X  (64 x ncols GEMM, K = 64) with V_WMMA_F32_16X16X4_F32.
// Block = 256 threads = 8 waves. M staged once in padded LDS. Each wave owns
// a distinct 16-column tile of X/Y and accumulates all 64 output rows in four
// 16x16 f32 accumulators (4 independent WMMAs per K-step hide XDL latency).
// X is read once and Y written once -> non-temporal hints on both.
//
// Templated on the column count: NC > 0 makes every row stride a literal so
// B-fragment addresses strength-reduce to immediate-offset loads / single
// literal adds instead of per-step v_mul_u64 chains. NC == 0 is the generic
// runtime fallback.
// ---------------------------------------------------------------------------
template <int NC>
__global__ __launch_bounds__(256) void givens_gemm_kernel(
    const float* __restrict__ M,
    const float* __restrict__ X,
    float* __restrict__ Y,
    int ncols_rt, int ntiles)
{
  const long ncols = (NC > 0) ? (long)NC : (long)ncols_rt;

  __shared__ float Msh[NPTS * LDS_STRIDE];

  const int tid = threadIdx.x;
  // Stage M (64x64) into LDS with padded row stride (coalesced b32 loads).
  for (int idx = tid; idx < NPTS * NPTS; idx += 256) {
    const int r = idx >> 6;
    const int c = idx & 63;
    Msh[r * LDS_STRIDE + c] = M[idx];
  }
  __syncthreads();

  const int wave = tid >> 5;
  const int lane = tid & 31;
  const int half = lane >> 4;  // half-wave selects K pair / M+8 rows
  const int ln   = lane & 15;  // N column within tile (B, C/D); M row (A)

  const int t = blockIdx.x * 8 + wave;
  if (t >= ntiles) return;
  const long c0 = (long)t * 16;

  v8f acc0 = {}, acc1 = {}, acc2 = {}, acc3 = {};

  // Per-lane base pointer for B loads: row (2*half), column c0+ln.
  const float* __restrict__ Xp = X + (long)(2 * half) * ncols + c0 + ln;
  // A fragment base: element (m = 16*mt + ln, k = kk + 2*half + vgpr)
  const int abase = ln * LDS_STRIDE + 2 * half;

#pragma unroll
  for (int kk = 0; kk < NPTS; kk += 4) {
    // B fragment (K x 16 tile of X): vgpr0 -> k = kk+2*half, vgpr1 -> +1.
    // Each half-wave load covers 64 contiguous bytes of one X row; with
    // constexpr ncols the +ncols byte offset fits the 24-bit imm field.
    v2f b;
    b.x = __builtin_nontemporal_load(Xp);
    b.y = __builtin_nontemporal_load(Xp + ncols);
    Xp += 4 * ncols;

    // A fragments for the 4 M-tiles (ds_load_b64, immediate offsets).
    const v2f a0 = *(const v2f*)&Msh[abase + 0 * 16 * LDS_STRIDE + kk];
    const v2f a1 = *(const v2f*)&Msh[abase + 1 * 16 * LDS_STRIDE + kk];
    const v2f a2 = *(const v2f*)&Msh[abase + 2 * 16 * LDS_STRIDE + kk];
    const v2f a3 = *(const v2f*)&Msh[abase + 3 * 16 * LDS_STRIDE + kk];

    acc0 = __builtin_amdgcn_wmma_f32_16x16x4_f32(false, a0, false, b,
                                                 (short)0, acc0, false, false);
    acc1 = __builtin_amdgcn_wmma_f32_16x16x4_f32(false, a1, false, b,
                                                 (short)0, acc1, false, false);
    acc2 = __builtin_amdgcn_wmma_f32_16x16x4_f32(false, a2, false, b,
                                                 (short)0, acc2, false, false);
    acc3 = __builtin_amdgcn_wmma_f32_16x16x4_f32(false, a3, false, b,
                                                 (short)0, acc3, false, false);
  }

  // C/D layout: vgpr v, half-wave h -> output row v + 8*h (+16 per m-tile),
  // column c0 + ln. Each half-wave store covers 64 contiguous bytes.
  const long col = c0 + ln;
#pragma unroll
  for (int v = 0; v < 8; ++v) {
    const long r = v + 8 * half;
    __builtin_nontemporal_store(acc0[v], &Y[(r +  0) * ncols + col]);
    __builtin_nontemporal_store(acc1[v], &Y[(r + 16) * ncols + col]);
    __builtin_nontemporal_store(acc2[v], &Y[(r + 32) * ncols + col]);
    __builtin_nontemporal_store(acc3[v], &Y[(r + 48) * ncols + col]);
  }
}

// ---------------------------------------------------------------------------
extern "C" void kernel_launch(void* const* d_in, const int* in_sizes, int n_in,
                              void* d_out, int out_size, void* d_ws, size_t ws_size,
                              hipStream_t stream) {
  (void)n_in; (void)out_size; (void)ws_size;

  const float* X      = (const float*)d_in[0];  // [64, ncols] f32
  const float* angles = (const float*)d_in[1];  // [2016] f32
  const float* mus    = (const float*)d_in[2];  // [64] f32
  float* Y = (float*)d_out;                     // [64, ncols] f32
  float* M = (float*)d_ws;                      // 64*64 f32 scratch (16 KB)

  const int ncols  = in_sizes[0] / NPTS;        // 500000 for this problem
  const int ntiles = ncols / 16;                // 16 | 500000 -> exact tiling

  fold_rotations_kernel<<<1, 256, 0, stream>>>(angles, mus, M);

  const int nblocks = (ntiles + 7) / 8;         // 8 waves/block, 1 tile/wave
  if (ncols == 500000) {
    givens_gemm_kernel<500000><<<nblocks, 256, 0, stream>>>(M, X, Y, ncols, ntiles);
  } else {
    givens_gemm_kernel<0><<<nblocks, 256, 0, stream>>>(M, X, Y, ncols, ntiles);
  }
}